// ConStealLoss_2422361555083
// MI455X (gfx1250) — compile-verified
//
#include <hip/hip_runtime.h>
#include <math.h>

typedef float v2f __attribute__((ext_vector_type(2)));
typedef float v8f __attribute__((ext_vector_type(8)));

#define BATCH   8192
#define DIM     128
#define INV_T   (1.0f / 0.07f)
#define NTILES  1024          // 2*BATCH/16 column tiles over [y; x]
#define LDSPAD  132           // 128 + 4 pad -> distinct banks across rows

// ---------------------------------------------------------------------------
// Kernel 1: fused exp-GEMM row sums.
// Block = 256 threads = 8 waves; each block owns a 16-row stripe of x.
// The 8 waves split the 1024 column tiles of the concatenated key matrix
// K = [y; x] (16384 rows). Each wave WMMA-computes 16x16 fp32 tiles of
// x @ K^T, applies exp(v/T), and accumulates per-row partial sums.
// S[i] = sum_j exp(<x_i,y_j>/T) + sum_j exp(<x_i,x_j>/T)   (incl. diagonals)
// ---------------------------------------------------------------------------
__global__ __launch_bounds__(256) void rowsum_expgemm_kernel(
    const float* __restrict__ x, const float* __restrict__ y,
    float* __restrict__ S)
{
    __shared__ float xs[16][LDSPAD];        // 16-row x stripe
    __shared__ float ks[8][16][LDSPAD];     // per-wave key tile
    __shared__ float partial[8][16];        // cross-wave row-sum combine

    const int tid  = threadIdx.x;
    const int w    = tid >> 5;              // wave id 0..7
    const int lane = tid & 31;
    const int ml   = lane & 15;             // M (for A) / N (for B) index
    const int hi   = lane >> 4;             // K-half select per fp32 layout
    const int i0   = blockIdx.x * 16;       // row stripe base

    // Stage the x stripe (16 x 128 fp32) into LDS, coalesced float4 loads.
    for (int idx = tid; idx < (16 * DIM) / 4; idx += 256) {
        int r  = idx >> 5;                  // 32 float4 per row
        int c4 = (idx & 31) * 4;
        float4 v = *(const float4*)(x + (size_t)(i0 + r) * DIM + c4);
        *(float4*)&xs[r][c4] = v;
    }
    __syncthreads();

    float rows[8];
#pragma unroll
    for (int r = 0; r < 8; ++r) rows[r] = 0.0f;

    for (int t = w; t < NTILES; t += 8) {
        const float* kb = (t < NTILES / 2) ? y : x;   // sim1 then sim2 tiles
        const int    j0 = (t & (NTILES / 2 - 1)) * 16;

        // Wave-cooperative stage of the 16x128 key tile: one full row
        // (512 contiguous bytes) per iteration -> perfectly coalesced.
#pragma unroll 4
        for (int rr = 0; rr < 16; ++rr) {
            float4 v = *(const float4*)(kb + (size_t)(j0 + rr) * DIM + lane * 4);
            *(float4*)&ks[w][rr][lane * 4] = v;
        }

        // 16x16 fp32 tile via 32 chained V_WMMA_F32_16X16X4_F32.
        // fp32 A layout: lanes 0-15 hold K = kk,kk+1 for row M=lane;
        //                lanes 16-31 hold K = kk+2,kk+3. B mirrors with N.
        v8f c = {};
#pragma unroll
        for (int kk = 0; kk < DIM; kk += 4) {
            v2f a = *(const v2f*)&xs[ml][kk + 2 * hi];
            v2f b = *(const v2f*)&ks[w][ml][kk + 2 * hi];
            c = __builtin_amdgcn_wmma_f32_16x16x4_f32(
                    false, a, false, b, (short)0, c, false, false);
        }

        // exp and fold the tile into per-row partial sums (trans ops
        // co-execute with the next tile's WMMA chain).
#pragma unroll
        for (int r = 0; r < 8; ++r)
            rows[r] += __expf(c[r] * INV_T);
    }

    // Sum over N: reduce across the 16 lanes of each half.
    // C layout: (vgpr r, lane l) = [M = r + 8*(l>>4)][N = l&15].
#pragma unroll
    for (int off = 8; off >= 1; off >>= 1) {
#pragma unroll
        for (int r = 0; r < 8; ++r)
            rows[r] += __shfl_xor(rows[r], off, 32);
    }
    if (lane == 0) {
#pragma unroll
        for (int r = 0; r < 8; ++r) partial[w][r] = rows[r];
    }
    if (lane == 16) {
#pragma unroll
        for (int r = 0; r < 8; ++r) partial[w][8 + r] = rows[r];
    }
    __syncthreads();

    if (tid < 16) {
        float s = 0.0f;
#pragma unroll
        for (int w2 = 0; w2 < 8; ++w2) s += partial[w2][tid];
        S[i0 + tid] = s;
    }
}

// ---------------------------------------------------------------------------
// Kernel 2: per-row diagonal dot products. One wave per row:
// dxy[i] = <x_i, y_i>,  dxx[i] = <x_i, x_i>.
// ---------------------------------------------------------------------------
__global__ __launch_bounds__(256) void diag_dots_kernel(
    const float* __restrict__ x, const float* __restrict__ y,
    float* __restrict__ dxy, float* __restrict__ dxx)
{
    const int gid  = blockIdx.x * blockDim.x + threadIdx.x;
    const int row  = gid >> 5;              // one wave32 per row
    const int lane = threadIdx.x & 31;
    if (row >= BATCH) return;

    float4 xa = *(const float4*)(x + (size_t)row * DIM + lane * 4);
    float4 ya = *(const float4*)(y + (size_t)row * DIM + lane * 4);

    float axy = xa.x * ya.x + xa.y * ya.y + xa.z * ya.z + xa.w * ya.w;
    float axx = xa.x * xa.x + xa.y * xa.y + xa.z * xa.z + xa.w * xa.w;

#pragma unroll
    for (int off = 16; off >= 1; off >>= 1) {
        axy += __shfl_xor(axy, off, 32);
        axx += __shfl_xor(axx, off, 32);
    }
    if (lane == 0) {
        dxy[row] = axy;
        dxx[row] = axx;
    }
}

// ---------------------------------------------------------------------------
// Kernel 3: finalize.
// loss = mean_i [ log(S_i - exp(dxy_i/T) - exp(dxx_i/T)) - dxy_i/T ]
// (since -log(l_pos/denom) = log(denom) - dxy/T).  Single block -> exact,
// deterministic reduction.
// ---------------------------------------------------------------------------
__global__ __launch_bounds__(1024) void finalize_kernel(
    const float* __restrict__ S, const float* __restrict__ dxy,
    const float* __restrict__ dxx, float* __restrict__ out)
{
    __shared__ float red[32];
    const int tid  = threadIdx.x;
    const int lane = tid & 31;
    const int w    = tid >> 5;

    float acc = 0.0f;
    for (int i = tid; i < BATCH; i += 1024) {
        float e1    = dxy[i] * INV_T;
        float lpos  = __expf(e1);
        float selfx = __expf(dxx[i] * INV_T);
        float denom = S[i] - lpos - selfx;   // l_neg_1 + l_neg_2
        acc += logf(denom) - e1;
    }
#pragma unroll
    for (int off = 16; off >= 1; off >>= 1)
        acc += __shfl_xor(acc, off, 32);
    if (lane == 0) red[w] = acc;
    __syncthreads();

    if (w == 0) {
        float v = red[lane];
#pragma unroll
        for (int off = 16; off >= 1; off >>= 1)
            v += __shfl_xor(v, off, 32);
        if (lane == 0) out[0] = v / (float)BATCH;
    }
}

// ---------------------------------------------------------------------------
extern "C" void kernel_launch(void* const* d_in, const int* in_sizes, int n_in,
                              void* d_out, int out_size, void* d_ws, size_t ws_size,
                              hipStream_t stream) {
    (void)in_sizes; (void)n_in; (void)out_size; (void)ws_size;
    const float* x = (const float*)d_in[0];
    const float* y = (const float*)d_in[1];

    // Workspace layout (96 KB): S[8192] | dxy[8192] | dxx[8192]
    float* S   = (float*)d_ws;
    float* dxy = S + BATCH;
    float* dxx = dxy + BATCH;

    rowsum_expgemm_kernel<<<BATCH / 16, 256, 0, stream>>>(x, y, S);
    diag_dots_kernel<<<(BATCH * 32) / 256, 256, 0, stream>>>(x, y, dxy, dxx);
    finalize_kernel<<<1, 1024, 0, stream>>>(S, dxy, dxx, (float*)d_out);
}